// SimpleMinsumMatcher_24962349924531
// MI455X (gfx1250) — compile-verified
//
#include <hip/hip_runtime.h>
#include <hip/hip_bf16.h>

// Problem constants (from setup_inputs): bs=8, nq=900, nc=91, T=200
#define BS   8
#define NQ   900
#define NC   91
#define NCP  96              // nc padded to multiple of 4 (WMMA K)
#define TT   200
#define QTOT (BS * NQ)       // 7200 rows of C
#define CTOT (BS * TT)       // 1600 cols of C
#define CELEMS ((size_t)QTOT * (size_t)CTOT)   // 11,520,000

typedef __attribute__((ext_vector_type(2))) float v2f;
typedef __attribute__((ext_vector_type(8))) float v8f;

// ---------------------------------------------------------------------------
// Kernel 1: focal class-cost table costdiff[q][c] = pos_cost - neg_cost,
// padded with zeros for c in [91,96).   (7200 x 96 floats, 2.76 MB -> L2)
// ---------------------------------------------------------------------------
__global__ __launch_bounds__(256) void prep_class(const float* __restrict__ logits,
                                                  float* __restrict__ costdiff) {
    int idx = blockIdx.x * blockDim.x + threadIdx.x;
    if (idx >= QTOT * NCP) return;
    int q = idx / NCP;
    int c = idx - q * NCP;
    float v = 0.0f;
    if (c < NC) {
        float x = logits[q * NC + c];
        float p = 1.0f / (1.0f + __expf(-x));
        float omp = 1.0f - p;
        float pos = 0.25f * omp * omp * (-__logf(p + 1e-8f));
        float neg = 0.75f * p * p * (-__logf(omp + 1e-8f));
        v = pos - neg;
    }
    costdiff[idx] = v;
}

// ---------------------------------------------------------------------------
// Kernel 2: one-hot of target ids, stored t-major: onehotT[t][c]
// (so a B fragment is a single 8-byte v2f load).  1600 x 96 floats.
// ---------------------------------------------------------------------------
__global__ __launch_bounds__(256) void prep_onehot(const int* __restrict__ ids,
                                                   float* __restrict__ onehotT) {
    int idx = blockIdx.x * blockDim.x + threadIdx.x;
    if (idx >= CTOT * NCP) return;
    int t = idx / NCP;
    int c = idx - t * NCP;
    onehotT[idx] = (c == ids[t]) ? 1.0f : 0.0f;
}

// ---------------------------------------------------------------------------
// Kernel 3: main cost kernel. One wave32 per 16x16 (q,t) tile.
//   cost_class tile via 24x V_WMMA_F32_16X16X4_F32 (bit-exact one-hot gather),
//   then per-lane L1 + GIoU epilogue on the accumulator layout.
// Grid: 45000 tiles / 8 waves per block = 5625 blocks x 256 threads.
// ---------------------------------------------------------------------------
__global__ __launch_bounds__(256)
void cost_main(const float* __restrict__ costdiff,   // [7200][96]
               const float* __restrict__ onehotT,    // [1600][96]
               const float* __restrict__ pred_boxes, // [7200][4] cxcywh
               const float* __restrict__ tgt_boxes,  // [1600][4] cxcywh
               float* __restrict__ C)                // [7200][1600]
{
    const int lane = threadIdx.x & 31;
    const int wave = threadIdx.x >> 5;
    const int tile = blockIdx.x * 8 + wave;      // 0..44999 (exact cover)
    const int qt = tile / 100;                   // 0..449 row tile
    const int tt = tile - qt * 100;              // 0..99  col tile
    const int q0 = qt * 16;
    const int t0 = tt * 16;

    const int half = lane >> 4;                  // lane half selects K pair
    const int l15  = lane & 15;

    // A 16x4 f32 layout: lanes 0-15 -> M=l15, VGPR{0,1} = K{2*half, 2*half+1}
    const float* Arow = costdiff + (size_t)(q0 + l15) * NCP + 2 * half;
    // B 4x16 f32 layout (mirrored): lane -> column n=l15, same K pairing
    const float* Brow = onehotT + (size_t)(t0 + l15) * NCP + 2 * half;

    v8f acc = {};
#pragma unroll
    for (int k = 0; k < NCP; k += 4) {
        v2f a = *(const v2f*)(Arow + k);
        v2f b = *(const v2f*)(Brow + k);
        // 8 args: (neg_a, A, neg_b, B, c_mod, C, reuse_a, reuse_b)
        acc = __builtin_amdgcn_wmma_f32_16x16x4_f32(
            false, a, false, b, (short)0, acc, false, false);
    }

    // --- epilogue: this lane owns column t = t0 + l15, rows m = r + 8*half ---
    const int t = t0 + l15;
    const float4 tb = ((const float4*)tgt_boxes)[t];
    const float tx0 = tb.x - 0.5f * tb.z, ty0 = tb.y - 0.5f * tb.w;
    const float tx1 = tb.x + 0.5f * tb.z, ty1 = tb.y + 0.5f * tb.w;
    const float area2 = (tx1 - tx0) * (ty1 - ty0);

#pragma unroll
    for (int r = 0; r < 8; ++r) {
        const int m = r + (half << 3);
        const int q = q0 + m;
        const float4 pb = ((const float4*)pred_boxes)[q];

        // L1 in cxcywh space
        float l1 = fabsf(pb.x - tb.x) + fabsf(pb.y - tb.y)
                 + fabsf(pb.z - tb.z) + fabsf(pb.w - tb.w);

        // GIoU in xyxy space
        float px0 = pb.x - 0.5f * pb.z, py0 = pb.y - 0.5f * pb.w;
        float px1 = pb.x + 0.5f * pb.z, py1 = pb.y + 0.5f * pb.w;
        float area1 = (px1 - px0) * (py1 - py0);
        float iw = fmaxf(fminf(px1, tx1) - fmaxf(px0, tx0), 0.0f);
        float ih = fmaxf(fminf(py1, ty1) - fmaxf(py0, ty0), 0.0f);
        float inter = iw * ih;
        float uni = area1 + area2 - inter;
        float iou = inter / uni;
        float cw = fmaxf(fmaxf(px1, tx1) - fminf(px0, tx0), 0.0f);
        float ch = fmaxf(fmaxf(py1, ty1) - fminf(py0, ty0), 0.0f);
        float areac = cw * ch;
        float giou = iou - (areac - uni) / areac;

        C[(size_t)q * CTOT + t] = acc[r] + l1 - giou;   // bbox + class - giou
    }
}

// ---------------------------------------------------------------------------
// Kernel 4: idx_i[b,t] = argmin over q (first occurrence) of the diagonal
// block C[b*900+q][b*200+t]. Coalesced: consecutive t -> consecutive addrs.
// ---------------------------------------------------------------------------
__global__ __launch_bounds__(256)
void argmin_diag(const float* __restrict__ C, void* __restrict__ outTail, int write64) {
    int b = blockIdx.x;           // 0..7
    int t = threadIdx.x;          // 0..255, active < 200
    if (t >= TT) return;
    const float* col = C + (size_t)(b * NQ) * CTOT + (size_t)(b * TT) + t;
    float best = col[0];
    int bestq = 0;
    for (int qq = 1; qq < NQ; ++qq) {
        float v = col[(size_t)qq * CTOT];
        if (v < best) { best = v; bestq = qq; }   // strict < : first min wins
    }
    if (write64) ((long long*)outTail)[b * TT + t] = (long long)bestq;
    else         ((int*)outTail)[b * TT + t]       = bestq;
}

// ---------------------------------------------------------------------------
extern "C" void kernel_launch(void* const* d_in, const int* in_sizes, int n_in,
                              void* d_out, int out_size, void* d_ws, size_t ws_size,
                              hipStream_t stream) {
    const float* logits = (const float*)d_in[0];   // (8,900,91) f32
    const float* pboxes = (const float*)d_in[1];   // (8,900,4)  f32
    const float* tboxes = (const float*)d_in[2];   // (8,200,4)  f32
    const int*   ids    = (const int*)d_in[3];     // (8,200)    i32

    float* C = (float*)d_out;

    float* costdiff = (float*)d_ws;                          // 7200*96 f32
    float* onehotT  = costdiff + (size_t)QTOT * NCP;         // 1600*96 f32

    {
        int n = QTOT * NCP;
        prep_class<<<(n + 255) / 256, 256, 0, stream>>>(logits, costdiff);
    }
    {
        int n = CTOT * NCP;
        prep_onehot<<<(n + 255) / 256, 256, 0, stream>>>(ids, onehotT);
    }

    // 45000 tiles, 8 waves (one tile each) per 256-thread block
    cost_main<<<45000 / 8, 256, 0, stream>>>(costdiff, onehotT, pboxes, tboxes, C);

    // idx_i (int64) appended after C in d_out; pick width from out_size
    long long tail = (long long)out_size - (long long)CELEMS;
    if (tail > 0) {
        void* tailPtr = (void*)(C + CELEMS);
        int w64 = (tail >= 2LL * BS * TT) ? 1 : 0;  // int64 counted as 2 f32 slots
        argmin_diag<<<BS, 256, 0, stream>>>(C, tailPtr, w64);
    }
}